// YOLOLoss_64905545777209
// MI455X (gfx1250) — compile-verified
//
#include <hip/hip_runtime.h>
#include <hip/hip_bf16.h>
#include <math.h>

#define S_GRID 7
#define NBB 2
#define NCLS 80
#define NBOX 20
#define RECLEN (NBB * (5 + NCLS))      // 170 floats per cell
#define BATCH 4096
#define NCELLS (BATCH * S_GRID * S_GRID)  // 200704
#define WAVES_PER_BLOCK 8
#define NBLK (NCELLS / WAVES_PER_BLOCK)   // 25088
#define LAMBDA_COORD 5.0f
#define LAMBDA_NOOBJ 0.5f

typedef float v2f __attribute__((ext_vector_type(2)));
typedef float v8f __attribute__((ext_vector_type(8)));

// Types for the async global->LDS builtin: param 0 is a 16-byte int vector
// pointer in the global address space (per hipcc's own diagnostic).
typedef int v4i __attribute__((vector_size(16)));
typedef __attribute__((address_space(1))) v4i* gv4i_ptr;
typedef __attribute__((address_space(3))) v4i* lv4i_ptr;

// ---------------------------------------------------------------------------
// Wave32 sum reduction via V_WMMA_F32_16X16X4_F32 (exact f32 adds).
// A-matrix layout (32-bit, 16x4): vgpr0 = {lanes0-15: K=0, lanes16-31: K=2},
// vgpr1 = {K=1 / K=3}. With a = {v, 0}, B = ones: D[m,n] = v[m] + v[m+16].
// D layout: vgpr j holds rows M=j (lanes0-15) / M=j+8 (lanes16-31), so
// s = sum(d[0..7]) gives half-sums; s + shfl_xor(s,16) = full wave sum on
// every lane.
// ---------------------------------------------------------------------------
__device__ __forceinline__ float wave_sum_wmma(float v) {
    v2f a; a.x = v;    a.y = 0.0f;
    v2f b; b.x = 1.0f; b.y = 1.0f;
    v8f c = {};
    c = __builtin_amdgcn_wmma_f32_16x16x4_f32(false, a, false, b,
                                              (short)0, c, false, false);
    float s = c[0] + c[1] + c[2] + c[3] + c[4] + c[5] + c[6] + c[7];
    s += __shfl_xor(s, 16, 32);
    return s;   // wave total, valid on all lanes
}

__device__ __forceinline__ float iou_xywh(float ax, float ay, float aw, float ah,
                                          float bx, float by, float bw, float bh) {
    float ax1 = ax - aw * 0.5f, ay1 = ay - ah * 0.5f;
    float ax2 = ax + aw * 0.5f, ay2 = ay + ah * 0.5f;
    float bx1 = bx - bw * 0.5f, by1 = by - bh * 0.5f;
    float bx2 = bx + bw * 0.5f, by2 = by + bh * 0.5f;
    float iw = fmaxf(fminf(ax2, bx2) - fmaxf(ax1, bx1), 0.0f);
    float ih = fmaxf(fminf(ay2, by2) - fmaxf(ay1, by1), 0.0f);
    float inter = iw * ih;
    float uni = (ax2 - ax1) * (ay2 - ay1) + (bx2 - bx1) * (by2 - by1) - inter;
    return inter / (uni + 1e-6f);
}

// ---------------------------------------------------------------------------
// Kernel 1: winner[cell] = NBOX (sentinel)
// ---------------------------------------------------------------------------
__global__ void init_winner_kernel(int* __restrict__ winner) {
    int i = blockIdx.x * blockDim.x + threadIdx.x;
    if (i < NCELLS) winner[i] = NBOX;
}

// ---------------------------------------------------------------------------
// Kernel 2: scatter boxes -> atomicMin on box order (matches .at[].min(order))
// ---------------------------------------------------------------------------
__global__ void scatter_boxes_kernel(const float* __restrict__ boxes,
                                     int* __restrict__ winner) {
    int t = blockIdx.x * blockDim.x + threadIdx.x;
    if (t >= BATCH * NBOX) return;
    int b = t / NBOX;
    int n = t - b * NBOX;
    const float* p = boxes + (size_t)t * 4;
    float x = (p[0] + p[2]) * 0.5f;
    float y = (p[1] + p[3]) * 0.5f;
    int j = min(max((int)floorf(x * (float)S_GRID), 0), S_GRID - 1);
    int i = min(max((int)floorf(y * (float)S_GRID), 0), S_GRID - 1);
    int cell = (b * S_GRID + i) * S_GRID + j;
    atomicMin(&winner[cell], n);
}

// ---------------------------------------------------------------------------
// Kernel 3: main loss. One wave32 per cell; per-block partial sum (no atomics).
// ---------------------------------------------------------------------------
__global__ __launch_bounds__(256)
void yolo_main_kernel(const float* __restrict__ pred,
                      const float* __restrict__ boxes,
                      const int* __restrict__ labels,
                      const int* __restrict__ winner,
                      float* __restrict__ partial) {
    const int wave = threadIdx.x >> 5;
    const int lane = threadIdx.x & 31;
    const int cell = blockIdx.x * WAVES_PER_BLOCK + wave;

    const float* rec = pred + (size_t)cell * RECLEN;
    __builtin_prefetch(rec, 0, 1);               // global_prefetch_b8

    const int win = winner[cell];                // wave-uniform
    float acc = 0.0f;

    const float p0c = rec[4];
    const float p1c = rec[85 + 4];

    if (win >= NBOX) {
        // no object: LAMBDA_NOOBJ * sum over both boxes of pconf^2
        if (lane == 0) acc = LAMBDA_NOOBJ * (p0c * p0c + p1c * p1c);
    } else {
        const int bidx = cell / (S_GRID * S_GRID);
        const float* tb = boxes + ((size_t)bidx * NBOX + win) * 4;
        const float x1 = tb[0], y1 = tb[1], x2 = tb[2], y2 = tb[3];
        const float tx = (x1 + x2) * 0.5f, ty = (y1 + y2) * 0.5f;
        const float tw = x2 - x1, th = y2 - y1;
        const float jf = fminf(fmaxf(floorf(tx * (float)S_GRID), 0.0f), (float)(S_GRID - 1));
        const float ifl = fminf(fmaxf(floorf(ty * (float)S_GRID), 0.0f), (float)(S_GRID - 1));
        const float xc = tx * (float)S_GRID - jf;
        const float yc = ty * (float)S_GRID - ifl;
        const int label = labels[bidx * NBOX + win];

        const float p0x = rec[0],  p0y = rec[1],  p0w = rec[2],  p0h = rec[3];
        const float p1x = rec[85], p1y = rec[86], p1w = rec[87], p1h = rec[88];

        const float iou0 = iou_xywh(p0x, p0y, p0w, p0h, xc, yc, tw, th);
        const float iou1 = iou_xywh(p1x, p1y, p1w, p1h, xc, yc, tw, th);
        const int best = (iou1 > iou0) ? 1 : 0;  // argmax: first max wins ties

        const float px = best ? p1x : p0x;
        const float py = best ? p1y : p0y;
        const float pw = best ? p1w : p0w;
        const float ph = best ? p1h : p0h;
        const float pc = best ? p1c : p0c;

        if (lane == 0) {
            float dx = px - xc, dy = py - yc;
            acc += LAMBDA_COORD * (dx * dx + dy * dy);
            float sw = sqrtf(fmaxf(pw, 1e-6f)) - sqrtf(fmaxf(tw, 1e-6f));
            float sh = sqrtf(fmaxf(ph, 1e-6f)) - sqrtf(fmaxf(th, 1e-6f));
            acc += LAMBDA_COORD * (sw * sw + sh * sh);
            float dc = pc - 1.0f;                // tconf == 1 for obj cells
            acc += dc * dc;
        }

        // class term of the responsible box only; lanes own classes
        // {lane, lane+32, lane+64}
        const float* cls = rec + best * 85 + 5;
        __builtin_prefetch(cls, 0, 1);
#pragma unroll
        for (int k = 0; k < 3; ++k) {
            int cI = lane + 32 * k;
            if (cI < NCLS) {
                float p = cls[cI];
                float t = (cI == label) ? 1.0f : 0.0f;
                float d = p - t;
                acc += d * d;
            }
        }
    }

    // EXEC all-ones here (branches reconverged) -> WMMA-based wave reduction
    float wsum = wave_sum_wmma(acc);

    __shared__ float wred[WAVES_PER_BLOCK];
    if (lane == 0) wred[wave] = wsum;
    __syncthreads();
    if (threadIdx.x == 0) {
        float t = 0.0f;
#pragma unroll
        for (int k = 0; k < WAVES_PER_BLOCK; ++k) t += wred[k];
        partial[blockIdx.x] = t;
    }
}

// ---------------------------------------------------------------------------
// Kernel 4: final reduction of NBLK partials. Stages tiles through LDS with
// async global->LDS copies (global_load_async_to_lds_b128 + s_wait_asynccnt).
// ---------------------------------------------------------------------------
__global__ __launch_bounds__(1024)
void final_reduce_kernel(const float* __restrict__ partial, float* __restrict__ out) {
    __shared__ float tile[4096];
    const int tid = threadIdx.x;
    float acc = 0.0f;

    for (int base = 0; base < NBLK; base += 4096) {
#pragma unroll
        for (int k = 0; k < 4; ++k) tile[tid + 1024 * k] = 0.0f;
        __syncthreads();

        const int idx = base + tid * 4;
#if __has_builtin(__builtin_amdgcn_global_load_async_to_lds_b128)
        if (idx + 3 < NBLK) {
            __builtin_amdgcn_global_load_async_to_lds_b128(
                (gv4i_ptr)(partial + idx),
                (lv4i_ptr)(&tile[tid * 4]),
                0, 0);
        } else {
            for (int k = 0; k < 4; ++k)
                if (idx + k < NBLK) tile[tid * 4 + k] = partial[idx + k];
        }
#if __has_builtin(__builtin_amdgcn_s_wait_asynccnt)
        __builtin_amdgcn_s_wait_asynccnt(0);
#else
        asm volatile("s_wait_asynccnt 0" ::: "memory");
#endif
#else
        for (int k = 0; k < 4; ++k)
            tile[tid * 4 + k] = (idx + k < NBLK) ? partial[idx + k] : 0.0f;
#endif
        __syncthreads();
#pragma unroll
        for (int k = 0; k < 4; ++k) acc += tile[tid + 1024 * k];
        __syncthreads();
    }

    float wsum = wave_sum_wmma(acc);

    __shared__ float wred[32];
    const int lane = tid & 31, wave = tid >> 5;
    if (lane == 0) wred[wave] = wsum;
    __syncthreads();
    if (tid == 0) {
        float t = 0.0f;
        for (int k = 0; k < 32; ++k) t += wred[k];
        out[0] = t * (1.0f / (float)BATCH);
    }
}

// ---------------------------------------------------------------------------
// Host launcher
// ---------------------------------------------------------------------------
extern "C" void kernel_launch(void* const* d_in, const int* in_sizes, int n_in,
                              void* d_out, int out_size, void* d_ws, size_t ws_size,
                              hipStream_t stream) {
    (void)in_sizes; (void)n_in; (void)out_size; (void)ws_size;
    const float* pred   = (const float*)d_in[0];
    const float* boxes  = (const float*)d_in[1];
    const int*   labels = (const int*)d_in[2];
    float* out = (float*)d_out;

    int*   winner  = (int*)d_ws;
    float* partial = (float*)((char*)d_ws + (size_t)NCELLS * sizeof(int));

    init_winner_kernel<<<(NCELLS + 255) / 256, 256, 0, stream>>>(winner);
    scatter_boxes_kernel<<<(BATCH * NBOX + 255) / 256, 256, 0, stream>>>(boxes, winner);
    yolo_main_kernel<<<NBLK, 256, 0, stream>>>(pred, boxes, labels, winner, partial);
    final_reduce_kernel<<<1, 1024, 0, stream>>>(partial, out);
}